// ChebyshevLayer_89489938580012
// MI455X (gfx1250) — compile-verified
//
#include <hip/hip_runtime.h>

#define PI_D 3.14159265358979323846

typedef float v2f __attribute__((ext_vector_type(2)));
typedef float v8f __attribute__((ext_vector_type(8)));

__device__ __forceinline__ v8f wmma4(v2f a, v2f b, v8f c) {
  // D = A(16x4,f32) * B(4x16,f32) + C(16x16,f32)
  return __builtin_amdgcn_wmma_f32_16x16x4_f32(false, a, false, b, (short)0, c, false, false);
}

// ---------------------------------------------------------------------------
// k_setup: K = Bi @ M   (Bi[n,k] = cos(pi n k / 255), derived from DCT-I)
//          Fw12[u][n] = forward-Chebyshev matrix rows u<12
// ---------------------------------------------------------------------------
__global__ __launch_bounds__(256) void k_setup(const float* __restrict__ Mmat,
                                               float* __restrict__ Kw,
                                               float* __restrict__ Fw) {
  __shared__ float cs[256];
  int x = blockIdx.x, t = threadIdx.x;
  cs[t] = (float)cos(PI_D * (double)x * (double)t / 255.0);
  __syncthreads();
  float acc = 0.f;
  for (int k = 0; k < 256; ++k) acc += cs[k] * Mmat[k * 256 + t];
  Kw[x * 256 + t] = acc;
  if (x == 0) {
    int n = t;
    float tn = (n == 0 || n == 255) ? 1.f : 2.f;
    for (int u = 0; u < 12; ++u) {
      float s = (u == 0) ? 0.5f : 1.f;
      Fw[u * 256 + n] = s * tn * (float)cos(PI_D * (double)u * (double)n / 255.0) / 255.f;
    }
  }
}

// ---------------------------------------------------------------------------
// k_phi: Phi = [ K @ M_1[:, :12]  |  K[:, :12] ]   (256 x 24, shared factor)
// ---------------------------------------------------------------------------
__global__ void k_phi(const float* __restrict__ Kw, const float* __restrict__ M1,
                      float* __restrict__ Phi) {
  int x = blockIdx.x, t = threadIdx.x;
  if (t < 12) {
    float acc = 0.f;
    for (int y = 0; y < 256; ++y) acc += Kw[x * 256 + y] * M1[y * 256 + t];
    Phi[x * 24 + t] = acc;
  } else if (t < 24) {
    Phi[x * 24 + t] = Kw[x * 256 + (t - 12)];
  }
}

// ---------------------------------------------------------------------------
// k_fwdA: T[bp,u,y,c] = sum_x Fw12[u,x] * X[bp,x,y,c]   (the 402MB read)
// ---------------------------------------------------------------------------
__global__ __launch_bounds__(256) void k_fwdA(const float* __restrict__ X,
                                              const float* __restrict__ Fw,
                                              float* __restrict__ T) {
  __shared__ float f[12 * 256];
  int t = threadIdx.x;
  for (int i = t; i < 12 * 256; i += 256) f[i] = Fw[i];
  __syncthreads();
  int bp = blockIdx.x >> 5;            // 0..47
  int y = ((blockIdx.x & 31) << 3) + (t >> 5);
  int c = t & 31;
  float acc[12];
#pragma unroll
  for (int u = 0; u < 12; ++u) acc[u] = 0.f;
  const float* Xp = X + ((size_t)bp * 256 * 256 + y) * 32 + c;
  for (int x = 0; x < 256; ++x) {
    float v = Xp[(size_t)x * 256 * 32];
#pragma unroll
    for (int u = 0; u < 12; ++u) acc[u] += f[u * 256 + x] * v;
  }
  float* Tp = T + ((size_t)bp * 12 * 256 + y) * 32 + c;
#pragma unroll
  for (int u = 0; u < 12; ++u) Tp[(size_t)u * 256 * 32] = acc[u];
}

// ---------------------------------------------------------------------------
// k_fwdB: xft[bp,uv,c] = sum_y T[bp,u,y,c] * Fw12[v,y]
// ---------------------------------------------------------------------------
__global__ __launch_bounds__(256) void k_fwdB(const float* __restrict__ T,
                                              const float* __restrict__ Fw,
                                              float* __restrict__ xft) {
  __shared__ float f[12 * 256];
  int t = threadIdx.x;
  for (int i = t; i < 12 * 256; i += 256) f[i] = Fw[i];
  __syncthreads();
  int id = blockIdx.x * 256 + t;       // 221184 total
  int c = id & 31;
  int v = (id >> 5) % 12;
  int u = (id / 384) % 12;
  int bp = id / 4608;
  const float* Tp = T + ((size_t)bp * 12 + u) * 256 * 32 + c;
  float acc = 0.f;
  for (int y = 0; y < 256; ++y) acc += Tp[(size_t)y * 32] * f[v * 256 + y];
  xft[((size_t)bp * 144 + u * 12 + v) * 32 + c] = acc;
}

// ---------------------------------------------------------------------------
// k_core (WMMA): per mode uv, core[bp,co] = xft[bp,ci] @ W[uv,ci,co]
//   M=48 (3 tiles), N=32 (2 tiles), K=32 (8 x wmma_f32_16x16x4_f32)
// ---------------------------------------------------------------------------
__global__ __launch_bounds__(32) void k_core(const float* __restrict__ xft,
                                             const float* __restrict__ Wt,
                                             float* __restrict__ core) {
  int uv = blockIdx.x / 6;
  int tl = blockIdx.x % 6;
  int m0 = (tl >> 1) * 16;
  int n0 = (tl & 1) * 16;
  int l = threadIdx.x;
  int lo = l & 15, hi = l >> 4;
  v8f acc = {0.f, 0.f, 0.f, 0.f, 0.f, 0.f, 0.f, 0.f};
  const float* Ap = xft + ((size_t)(m0 + lo) * 144 + uv) * 32;
  const float* Bp = Wt + (size_t)uv * 32 * 32 + n0 + lo;
#pragma unroll
  for (int k0 = 0; k0 < 32; k0 += 4) {
    v2f a, b;
    a.x = Ap[k0 + 2 * hi];
    a.y = Ap[k0 + 2 * hi + 1];
    b.x = Bp[(size_t)(k0 + 2 * hi) * 32];
    b.y = Bp[(size_t)(k0 + 2 * hi + 1) * 32];
    acc = wmma4(a, b, acc);
  }
#pragma unroll
  for (int r = 0; r < 8; ++r) {
    int m = m0 + r + 8 * hi;
    core[((size_t)m * 144 + uv) * 32 + n0 + lo] = acc[r];
  }
}

// ---------------------------------------------------------------------------
// k_S: assemble per-slice 24x24 S from core + BC/continuity corrections.
//   D_p = A C_p A^T, A = M_1[:, :12]; every correction = H*u + K12*w.
// ---------------------------------------------------------------------------
__global__ __launch_bounds__(256) void k_S(const float* __restrict__ core,
                                           const float* __restrict__ M1,
                                           float* __restrict__ Sall) {
  __shared__ float A12[12][12];        // A12[x][v] = M_1[x, v], x,v < 12
  int t = threadIdx.x;
  if (t < 144) A12[t / 12][t % 12] = M1[(t / 12) * 256 + (t % 12)];
  __syncthreads();
  int id = blockIdx.x * 256 + t;       // 0..511 = (b, co)
  int b = id >> 5, co = id & 31;
  const float* Cb = core + (size_t)b * 3 * 144 * 32 + co;
  auto C = [&](int p, int u, int v) -> float {
    return Cb[((size_t)p * 144 + u * 12 + v) * 32];
  };

  // cross-patch vectors: h1_0 = C0 a1, h0_1 = C1 a0, g1_1 = C1^T a1, g0_2 = C2^T a0
  float h1_0[12], h0_1[12], g1_1[12], g0_2[12];
  for (int u = 0; u < 12; ++u) {
    float s0 = 0.f, s1 = 0.f;
    for (int v = 0; v < 12; ++v) { s0 += C(0, u, v) * A12[1][v]; s1 += C(1, u, v) * A12[0][v]; }
    h1_0[u] = s0; h0_1[u] = s1;
  }
  for (int v = 0; v < 12; ++v) {
    float s1 = 0.f, s2 = 0.f;
    for (int u = 0; u < 12; ++u) { s1 += C(1, u, v) * A12[1][u]; s2 += C(2, u, v) * A12[0][u]; }
    g1_1[v] = s1; g0_2[v] = s2;
  }
  float tmp12 = 0.f;
  for (int v = 0; v < 12; ++v) tmp12 += A12[1][v] * h0_1[v];
  tmp12 *= (1.f / 3.f);
  float tmp1[12], tmp2[12];
  for (int x = 0; x < 12; ++x) {
    float d0 = 0.f, d1 = 0.f, e1 = 0.f, e2 = 0.f;
    for (int v = 0; v < 12; ++v) {
      d0 += A12[x][v] * h1_0[v];
      d1 += A12[x][v] * h0_1[v];
      e1 += A12[x][v] * g1_1[v];
      e2 += A12[x][v] * g0_2[v];
    }
    tmp1[x] = 0.5f * (d0 + d1);
    tmp2[x] = 0.5f * (e2 + e1);
  }
  tmp1[0] = 0.f; tmp1[1] = tmp12;
  tmp2[0] = tmp12; tmp2[1] = 0.f;

  for (int p = 0; p < 3; ++p) {
    float g0[12], g1[12], h0[12], h1[12];
    for (int v = 0; v < 12; ++v) {
      float s0 = 0.f, s1 = 0.f;
      for (int u = 0; u < 12; ++u) { float cv = C(p, u, v); s0 += cv * A12[0][u]; s1 += cv * A12[1][u]; }
      g0[v] = s0; g1[v] = s1;
    }
    for (int u = 0; u < 12; ++u) {
      float s0 = 0.f, s1 = 0.f;
      for (int v = 0; v < 12; ++v) { float cv = C(p, u, v); s0 += cv * A12[0][v]; s1 += cv * A12[1][v]; }
      h0[u] = s0; h1[u] = s1;
    }
    float D00 = 0.f, D01 = 0.f, D10 = 0.f, D11 = 0.f;
    for (int u = 0; u < 12; ++u) {
      D00 += A12[0][u] * h0[u]; D10 += A12[1][u] * h0[u];
      D01 += A12[0][u] * h1[u]; D11 += A12[1][u] * h1[u];
    }
    float uR0[12], uR1[12], uC0[12], uC1[12], wR0[12], wR1[12], wC0[12], wC1[12];
    for (int i = 0; i < 12; ++i)
      uR0[i] = uR1[i] = uC0[i] = uC1[i] = wR0[i] = wR1[i] = wC0[i] = wC1[i] = 0.f;
    if (p == 0) {
      for (int i = 0; i < 12; ++i) { uR0[i] = -g0[i]; uR1[i] = -g1[i]; uC0[i] = -h0[i]; }
      wR1[1] = tmp12;
      wC0[0] = D00; wC0[1] = D10;
      for (int x = 2; x < 12; ++x) {
        float d = 0.f;
        for (int v = 0; v < 12; ++v) d += A12[x][v] * h1[v];
        wC1[x] = tmp1[x] - d;
      }
    } else if (p == 1) {
      for (int i = 0; i < 12; ++i) { uR0[i] = -g0[i]; uC1[i] = -h1[i]; }
      for (int y = 0; y < 12; ++y) {
        float d = 0.f;
        for (int v = 0; v < 12; ++v) d += A12[y][v] * g1[v];
        wR1[y] = tmp2[y] - d;
      }
      for (int x = 2; x < 12; ++x) {
        float d = 0.f;
        for (int v = 0; v < 12; ++v) d += A12[x][v] * h0[v];
        wC0[x] = tmp1[x] - d;
      }
      wC1[0] = D01; wC1[1] = D11;
    } else {
      for (int i = 0; i < 12; ++i) { uR1[i] = -g1[i]; uC0[i] = -h0[i]; uC1[i] = -h1[i]; }
      for (int y = 0; y < 12; ++y) {
        float d = 0.f;
        for (int v = 0; v < 12; ++v) d += A12[y][v] * g0[v];
        wR0[y] = tmp2[y] - d;
      }
      wC0[0] = D00; wC0[1] = D10;
      wC1[0] = D01; wC1[1] = D11;
    }
    float* S = Sall + ((size_t)(b * 3 + p) * 32 + co) * 576;
    for (int r = 0; r < 24; ++r) {
      for (int cn = 0; cn < 24; ++cn) {
        float val;
        if (r < 12) {
          if (cn < 12) val = C(p, r, cn);
          else if (cn == 12) val = uC0[r];
          else if (cn == 13) val = uC1[r];
          else val = 0.f;
        } else if (r == 12) {
          if (cn < 12) val = uR0[cn];
          else { val = wR0[cn - 12]; if (cn == 12) val += wC0[0]; if (cn == 13) val += wC1[0]; }
        } else if (r == 13) {
          if (cn < 12) val = uR1[cn];
          else { val = wR1[cn - 12]; if (cn == 12) val += wC0[1]; if (cn == 13) val += wC1[1]; }
        } else {
          if (cn == 12) val = wC0[r - 12];
          else if (cn == 13) val = wC1[r - 12];
          else val = 0.f;
        }
        S[r * 24 + cn] = val;
      }
    }
  }
}

// ---------------------------------------------------------------------------
// k_W: W_s = Phi @ S_s   (256 x 24 per slice)
// ---------------------------------------------------------------------------
__global__ __launch_bounds__(256) void k_W(const float* __restrict__ Sall,
                                           const float* __restrict__ Phi,
                                           float* __restrict__ Wall) {
  __shared__ float S[576];
  int t = threadIdx.x;
  size_t s = blockIdx.x;
  for (int i = t; i < 576; i += 256) S[i] = Sall[s * 576 + i];
  __syncthreads();
  float ph[24], o[24];
#pragma unroll
  for (int k = 0; k < 24; ++k) ph[k] = Phi[t * 24 + k];
#pragma unroll
  for (int n = 0; n < 24; ++n) o[n] = 0.f;
#pragma unroll
  for (int k = 0; k < 24; ++k) {
    float p = ph[k];
#pragma unroll
    for (int n = 0; n < 24; ++n) o[n] += p * S[k * 24 + n];
  }
  float* Wp = Wall + s * (256 * 24) + t * 24;
#pragma unroll
  for (int n = 0; n < 24; ++n) Wp[n] = o[n];
}

// ---------------------------------------------------------------------------
// k_final (WMMA): V = W_s @ Phi^T, per (bp, xtile, ytile) block over all 32
// channels; stage 16x16x32 tile in LDS, then fully-coalesced float4 stores.
// ---------------------------------------------------------------------------
__global__ __launch_bounds__(256) void k_final(const float* __restrict__ Wall,
                                               const float* __restrict__ Phi,
                                               float* __restrict__ out) {
  __shared__ __align__(16) float phiB[16 * 24];
  __shared__ __align__(16) float tile[16 * 16 * 32];
  int t = threadIdx.x;
  int blk = blockIdx.x;
  int bp = blk >> 8;
  int x0 = ((blk >> 4) & 15) * 16;
  int y0 = (blk & 15) * 16;
  for (int i = t; i < 16 * 24; i += 256) phiB[i] = Phi[(y0 + i / 24) * 24 + (i % 24)];
  __syncthreads();
  int wave = t >> 5, lane = t & 31, lo = lane & 15, hi = lane >> 4;
  for (int cc = 0; cc < 4; ++cc) {
    int c = wave * 4 + cc;
    const float* Wp = Wall + ((size_t)bp * 32 + c) * (256 * 24) + (x0 + lo) * 24;
    v8f acc = {0.f, 0.f, 0.f, 0.f, 0.f, 0.f, 0.f, 0.f};
#pragma unroll
    for (int k0 = 0; k0 < 24; k0 += 4) {
      v2f a, b;
      a.x = Wp[k0 + 2 * hi];
      a.y = Wp[k0 + 2 * hi + 1];
      b.x = phiB[lo * 24 + k0 + 2 * hi];
      b.y = phiB[lo * 24 + k0 + 2 * hi + 1];
      acc = wmma4(a, b, acc);
    }
#pragma unroll
    for (int r = 0; r < 8; ++r) {
      int m = r + 8 * hi;
      tile[(m * 16 + lo) * 32 + c] = acc[r];
    }
  }
  __syncthreads();
  const float4* tp = (const float4*)tile;
  float4* op = (float4*)(out + (((size_t)bp * 256 + x0) * 256 + y0) * 32);
  for (int i = t; i < 16 * 128; i += 256) {
    int m = i >> 7, q = i & 127;
    op[(size_t)m * 2048 + q] = tp[m * 128 + q];
  }
}

// ---------------------------------------------------------------------------
extern "C" void kernel_launch(void* const* d_in, const int* in_sizes, int n_in,
                              void* d_out, int out_size, void* d_ws, size_t ws_size,
                              hipStream_t stream) {
  const float* X  = (const float*)d_in[0];   // (16,3,256,256,32)
  const float* Wt = (const float*)d_in[1];   // (12,12,32,32)
  const float* M  = (const float*)d_in[2];   // (256,256)
  const float* M1 = (const float*)d_in[3];   // (256,256)
  float* out = (float*)d_out;
  float* ws = (float*)d_ws;

  float* Fw   = ws;                  // 12*256        = 3072
  float* Kw   = Fw + 3072;           // 256*256       = 65536
  float* Phi  = Kw + 65536;          // 256*24        = 6144
  float* T    = Phi + 6144;          // 48*12*256*32  = 4718592
  float* xft  = T + 4718592;         // 48*144*32     = 221184
  float* core = xft + 221184;        // 48*144*32     = 221184
  float* Sall = core + 221184;       // 1536*576      = 884736
  float* Wall = Sall + 884736;       // 1536*256*24   = 9437184

  k_setup<<<dim3(256), dim3(256), 0, stream>>>(M, Kw, Fw);
  k_phi  <<<dim3(256), dim3(32),  0, stream>>>(Kw, M1, Phi);
  k_fwdA <<<dim3(1536), dim3(256), 0, stream>>>(X, Fw, T);
  k_fwdB <<<dim3(864), dim3(256), 0, stream>>>(T, Fw, xft);
  k_core <<<dim3(864), dim3(32),  0, stream>>>(xft, Wt, core);
  k_S    <<<dim3(2), dim3(256),   0, stream>>>(core, M1, Sall);
  k_W    <<<dim3(1536), dim3(256), 0, stream>>>(Sall, Phi, Wall);
  k_final<<<dim3(12288), dim3(256), 0, stream>>>(Wall, Phi, out);
}